// Model_60352880444046
// MI455X (gfx1250) — compile-verified
//
#include <hip/hip_runtime.h>
#include <hip/hip_bf16.h>
#include <stdint.h>

typedef float v2f __attribute__((ext_vector_type(2)));
typedef float v8f __attribute__((ext_vector_type(8)));
typedef unsigned int v4u __attribute__((ext_vector_type(4)));
typedef int v4i __attribute__((ext_vector_type(4)));
typedef int v8i __attribute__((ext_vector_type(8)));

#define NB 4096
#define NT 256
#define NF 32
#define NH 6
#define NG 24        // 4*H
#define LROW 36      // padded LDS row stride in floats (32 data + 4 pad, via TDM pad)

// ds_swizzle_b32 helpers (group-of-32 mode): src_lane = ((lane & AND) | OR) ^ XOR
#define SWZ_XOR(m)   (0x7C00 | (m))                 // and=0x1f, or=0, xor=m
#define SWZ_GRP8(o)  ((0x07 << 10) | ((o) << 5))    // and=7, or=o : broadcast from group
template <int OFS>
__device__ __forceinline__ float swz(float v) {
  return __int_as_float(__builtin_amdgcn_ds_swizzle(__float_as_int(v), OFS));
}

// DPP16 lane permute (VALU, no LDS)
template <int CTRL>
__device__ __forceinline__ float dppf(float v) {
  int i = __float_as_int(v);
  return __int_as_float(__builtin_amdgcn_update_dpp(i, i, CTRL, 0xF, 0xF, false));
}
// 16-lane sum reduction via DPP involutions (progressive uniformity)
__device__ __forceinline__ float row16_sum(float s) {
  s += dppf<0xB1>(s);    // quad_perm xor 1
  s += dppf<0x4E>(s);    // quad_perm xor 2
  s += dppf<0x141>(s);   // row_half_mirror (xor4-equivalent on quad-uniform data)
  s += dppf<0x140>(s);   // row_mirror (xor8-equivalent on 8-uniform data)
  return s;
}

__device__ __forceinline__ float fast_rcp(float v) { return __builtin_amdgcn_rcpf(v); }

// ---------------------------------------------------------------------------
// K1: per 16-row tile (fixed t, 16 consecutive b):
//   x tile staged via TDM (tensor_load_to_lds, HW-padded rows: 32dw + 4dw pad)
//   scores = x@Wa + ba  (WMMA f32 16x16x4, K=32)
//   w = softmax(scores) over 32 features (C-layout, DPP row reductions)
//   xz = (x*w)@Wl + bl  (WMMA, feat A-frags = xA ⊙ wA)
//   store xz to ws in [T,B,24] layout
// ---------------------------------------------------------------------------
__global__ __launch_bounds__(256) void k1_attn_proj(
    const float* __restrict__ x,
    const float* __restrict__ Wa, const float* __restrict__ ba,
    const float* __restrict__ Wl, const float* __restrict__ bl,
    float* __restrict__ xz)
{
  __shared__ __align__(16) float lds[8][2][16 * LROW];

  const int tid = threadIdx.x;
  const int wv  = tid >> 5;
  const int ln  = tid & 31;
  const int hl  = ln >> 4;
  const int l15 = ln & 15;

  const int gw = blockIdx.x * 8 + wv;    // 0..65535
  const int bg = gw & (NB / 16 - 1);
  const int t  = gw >> 8;

  float* ldsX = &lds[wv][0][0];
  float* ldsW = &lds[wv][1][0];

  // --- TDM: DMA the 16x32 f32 tile (row stride NT*NF elems) into LDS ------
  // D# group0: count=1 | lds_addr | global_addr(57b) | type=2
  // D# group1: data_size=4B, pad_enable, pad_interval=32dw, pad_amount=4dw,
  //            tensor 32x16, tile 32x16, dim0_stride=8192
  {
    const uint64_t ga  = (uint64_t)(uintptr_t)(x + ((size_t)(bg * 16)) * NT * NF + (size_t)t * NF);
    const uint32_t lof = (uint32_t)(uintptr_t)ldsX;
    v4u g0;
    g0.x = 1u;                                     // count = 1 valid descriptor
    g0.y = lof;                                    // lds_addr (bytes)
    g0.z = (uint32_t)ga;                           // global_addr[31:0]
    g0.w = (uint32_t)(ga >> 32) | 0x80000000u;     // global_addr[56:32] | type=2
    v8i g1;
    g1[0] = (int)((2u << 16)                       // data_size = 4 bytes
                | (1u << 20)                       // pad_enable
                | (4u << 22)                       // pad_interval: 32 DWORDs
                | (3u << 25));                     // pad_amount: 4 DWORDs
    g1[1] = (int)(32u << 16);                      // tensor_dim0 = 32
    g1[2] = (int)(16u << 16);                      // tensor_dim1 = 16
    g1[3] = (int)(32u << 16);                      // tile_dim0 = 32
    g1[4] = 16;                                    // tile_dim1 = 16 (tile_dim2 = 0)
    g1[5] = NT * NF;                               // tensor_dim0_stride = 8192
    g1[6] = 0;
    g1[7] = 0;
    const v4i gz4 = {0, 0, 0, 0};
    const v8i gz8 = {0, 0, 0, 0, 0, 0, 0, 0};
    __builtin_amdgcn_tensor_load_to_lds(g0, g1, gz4, gz4, gz8, 0);
  }

  // --- B fragments for Wa (32x32) and Wl (32x24), overlapped with the DMA -
  v2f aB[8][2], lB[8][2];
#pragma unroll
  for (int kb = 0; kb < 8; ++kb) {
    const int k0 = 4 * kb + hl, k1 = k0 + 2;
#pragma unroll
    for (int n = 0; n < 2; ++n) {
      const int col = n * 16 + l15;
      v2f va; va.x = Wa[k0 * NF + col]; va.y = Wa[k1 * NF + col];
      aB[kb][n] = va;
      v2f vl;
      vl.x = (col < NG) ? Wl[k0 * NG + col] : 0.0f;
      vl.y = (col < NG) ? Wl[k1 * NG + col] : 0.0f;
      lB[kb][n] = vl;
    }
  }

  v8f s0, s1, z0, z1;
  {
    const float ba0 = ba[l15], ba1 = ba[16 + l15];
    const float bl0 = bl[l15];
    const float bl1 = (l15 < 8) ? bl[16 + l15] : 0.0f;
#pragma unroll
    for (int r = 0; r < 8; ++r) { s0[r] = ba0; s1[r] = ba1; z0[r] = bl0; z1[r] = bl1; }
  }

  asm volatile("s_wait_tensorcnt 0" ::: "memory");

  // --- A fragments of x: lane l15 = row, K = 4*kb + 2*hl ------------------
  v2f xA[8];
#pragma unroll
  for (int kb = 0; kb < 8; ++kb)
    xA[kb] = *(const v2f*)&ldsX[l15 * LROW + 4 * kb + 2 * hl];

  // --- GEMM1: scores = x @ Wa + ba ----------------------------------------
#pragma unroll
  for (int kb = 0; kb < 8; ++kb) {
    s0 = __builtin_amdgcn_wmma_f32_16x16x4_f32(false, xA[kb], false, aB[kb][0], (short)0, s0, false, false);
    s1 = __builtin_amdgcn_wmma_f32_16x16x4_f32(false, xA[kb], false, aB[kb][1], (short)0, s1, false, false);
  }

  // --- softmax over 32 features per row (no max pass: scores are O(10)) ---
  v8f w0, w1;
  float ssum[8];
#pragma unroll
  for (int r = 0; r < 8; ++r) {
    float ea = __expf(s0[r]);
    float eb = __expf(s1[r]);
    w0[r] = ea; w1[r] = eb;
    ssum[r] = ea + eb;
  }
#pragma unroll
  for (int r = 0; r < 8; ++r) {
    float inv = fast_rcp(row16_sum(ssum[r]));
    w0[r] *= inv;
    w1[r] *= inv;
  }

  // --- weights tile C-layout -> LDS (re-read in A-layout) -----------------
#pragma unroll
  for (int r = 0; r < 8; ++r) {
    const int row = r + 8 * hl;
    ldsW[row * LROW + l15]      = w0[r];
    ldsW[row * LROW + 16 + l15] = w1[r];
  }

  // --- GEMM2: xz = (x*w) @ Wl + bl ----------------------------------------
#pragma unroll
  for (int kb = 0; kb < 8; ++kb) {
    v2f wA = *(const v2f*)&ldsW[l15 * LROW + 4 * kb + 2 * hl];
    v2f fA; fA.x = xA[kb].x * wA.x; fA.y = xA[kb].y * wA.y;
    z0 = __builtin_amdgcn_wmma_f32_16x16x4_f32(false, fA, false, lB[kb][0], (short)0, z0, false, false);
    z1 = __builtin_amdgcn_wmma_f32_16x16x4_f32(false, fA, false, lB[kb][1], (short)0, z1, false, false);
  }

  // --- store xz tile -> ws [T, B, 24] -------------------------------------
  float* dst = xz + ((size_t)t * NB + (size_t)bg * 16) * NG;
#pragma unroll
  for (int r = 0; r < 8; ++r) {
    const int row = r + 8 * hl;
    dst[row * NG + l15] = z0[r];
    if (l15 < 8) dst[row * NG + 16 + l15] = z1[r];
  }
}

// ---------------------------------------------------------------------------
// K2: sequential LSTM scan + fused head.
// Wave = 8 batch rows x 4 gate groups (q=0:i, 1:f, 2:g(softmax), 3:o).
// W10 staged via async-to-LDS. Each lane: 6x6 slice of h@Ul, its group's
// nonlinearity (shared exp(z) feeds sigmoid and softmax), ds_swizzle
// group-of-8 broadcasts gather i/f/g/o. y split 3 cols/group.
// xz loads software-pipelined one step ahead + prefetch two ahead.
// ---------------------------------------------------------------------------
__global__ __launch_bounds__(128) void k2_lstm_tail(
    const float* __restrict__ xz,
    const float* __restrict__ Ul,
    const float* __restrict__ W10, const float* __restrict__ b10,
    const float* __restrict__ Wo,  const float* __restrict__ bo,
    float* __restrict__ out)
{
  __shared__ __align__(16) float sW10[NT * NH * 10];   // 61440 B
  {
    const uint32_t dofs = (uint32_t)(uintptr_t)&sW10[threadIdx.x * 4];
    const float* src = W10 + threadIdx.x * 4;
#pragma unroll
    for (int it = 0; it < 30; ++it) {
      asm volatile("global_load_async_to_lds_b128 %0, %1, off"
                   :: "v"(dofs + 2048u * it), "v"(src + 512 * it) : "memory");
    }
    asm volatile("s_wait_asynccnt 0" ::: "memory");
  }
  __syncthreads();

  const int ln = threadIdx.x & 31;
  const int wv = threadIdx.x >> 5;
  const int l8 = ln & 7;
  const int q  = ln >> 3;                    // gate group
  const int b  = (blockIdx.x * 4 + wv) * 8 + l8;
  const int j0 = q * 6;                      // this lane's gate-channel base

  float U[6][6];
#pragma unroll
  for (int k = 0; k < 6; ++k)
#pragma unroll
    for (int j = 0; j < 6; ++j)
      U[k][j] = Ul[k * NG + j0 + j];

  // y columns handled by this group: ybase..ybase+2 (clamped, masked at end)
  const int ybase = q * 3;
  int ycol[3]; float ymask[3];
#pragma unroll
  for (int j = 0; j < 3; ++j) {
    const int cc = ybase + j;
    ycol[j]  = (cc < 10) ? cc : 9;
    ymask[j] = (cc < 10) ? 1.0f : 0.0f;
  }

  float h[6], c[6], y[3];
#pragma unroll
  for (int k = 0; k < 6; ++k) { h[k] = 0.0f; c[k] = 0.0f; }
#pragma unroll
  for (int j = 0; j < 3; ++j) y[j] = 0.0f;

  const float* zbase = xz + (size_t)b * NG + j0;
  float2 nz0 = *(const float2*)(zbase);
  float2 nz1 = *(const float2*)(zbase + 2);
  float2 nz2 = *(const float2*)(zbase + 4);

  for (int t = 0; t < NT; ++t) {
    float z[6] = { nz0.x, nz0.y, nz1.x, nz1.y, nz2.x, nz2.y };

    // pipeline: issue next step's loads now, prefetch two ahead
    const int tn = (t < NT - 1) ? t + 1 : t;
    const float* zpn = zbase + (size_t)tn * (NB * NG);
    nz0 = *(const float2*)(zpn);
    nz1 = *(const float2*)(zpn + 2);
    nz2 = *(const float2*)(zpn + 4);
    const int tp = (t < NT - 2) ? t + 2 : t;
    __builtin_prefetch(zbase + (size_t)tp * (NB * NG), 0, 1);

    // recurrence matvec: 6x6 slice per lane
#pragma unroll
    for (int k = 0; k < 6; ++k)
#pragma unroll
      for (int j = 0; j < 6; ++j)
        z[j] = fmaf(h[k], U[k][j], z[j]);

    // group nonlinearity: shared exp(z) feeds sigmoid and softmax
    float e[6]; float ss = 0.0f;
#pragma unroll
    for (int k = 0; k < 6; ++k) { e[k] = __expf(z[k]); ss += e[k]; }
    const float invs = fast_rcp(ss);
    float a[6];
#pragma unroll
    for (int k = 0; k < 6; ++k) {
      const float sg = e[k] * fast_rcp(1.0f + e[k]);   // sigmoid(z)
      const float sv = e[k] * invs;                    // softmax(z)
      a[k] = (q == 2) ? sv : sg;
    }

    // gather all four gates via group-of-8 swizzle broadcasts (independent)
    float i_[6], f_[6], g_[6], o_[6];
#pragma unroll
    for (int k = 0; k < 6; ++k) {
      i_[k] = swz<SWZ_GRP8(0)>(a[k]);
      f_[k] = swz<SWZ_GRP8(8)>(a[k]);
      g_[k] = swz<SWZ_GRP8(16)>(a[k]);
      o_[k] = swz<SWZ_GRP8(24)>(a[k]);
    }
#pragma unroll
    for (int k = 0; k < 6; ++k) c[k] = fmaf(f_[k], c[k], i_[k] * g_[k]);

    // h = o * softmax(c)   (keep max pass: c can reach O(T))
    float mc = c[0];
#pragma unroll
    for (int k = 1; k < 6; ++k) mc = fmaxf(mc, c[k]);
    float ec[6]; float sc = 0.0f;
#pragma unroll
    for (int k = 0; k < 6; ++k) { ec[k] = __expf(c[k] - mc); sc += ec[k]; }
    const float invc = fast_rcp(sc);
#pragma unroll
    for (int k = 0; k < 6; ++k) h[k] = o_[k] * (ec[k] * invc);

    // y accumulation (off the critical path)
    const float* wr = &sW10[t * 60];
#pragma unroll
    for (int k = 0; k < 6; ++k)
#pragma unroll
      for (int j = 0; j < 3; ++j)
        y[j] = fmaf(h[k], wr[k * 10 + ycol[j]], y[j]);
  }

  // head: mask + bias, out = y@Wo reduced across the 4 gate groups
#pragma unroll
  for (int j = 0; j < 3; ++j) y[j] = ymask[j] * (y[j] + b10[ycol[j]]);

  float po[8];
#pragma unroll
  for (int jo = 0; jo < 8; ++jo) {
    float acc = 0.0f;
#pragma unroll
    for (int j = 0; j < 3; ++j) acc = fmaf(y[j], Wo[ycol[j] * 8 + jo], acc);
    po[jo] = acc;
  }
#pragma unroll
  for (int jo = 0; jo < 8; ++jo) {
    po[jo] += swz<SWZ_XOR(8)>(po[jo]);
    po[jo] += swz<SWZ_XOR(16)>(po[jo]);
    po[jo] += bo[jo];
  }
  if (q == 0) {
#pragma unroll
    for (int jo = 0; jo < 8; ++jo) out[(size_t)b * 8 + jo] = po[jo];
  }
}

// ---------------------------------------------------------------------------
extern "C" void kernel_launch(void* const* d_in, const int* in_sizes, int n_in,
                              void* d_out, int out_size, void* d_ws, size_t ws_size,
                              hipStream_t stream) {
  (void)in_sizes; (void)n_in; (void)out_size; (void)ws_size;
  const float* x   = (const float*)d_in[0];
  const float* Wa  = (const float*)d_in[1];
  const float* ba  = (const float*)d_in[2];
  const float* Wl  = (const float*)d_in[3];
  const float* Ul  = (const float*)d_in[4];
  const float* bl  = (const float*)d_in[5];
  const float* W10 = (const float*)d_in[6];
  const float* b10 = (const float*)d_in[7];
  const float* Wo  = (const float*)d_in[8];
  const float* bo  = (const float*)d_in[9];
  float* out = (float*)d_out;
  float* xz  = (float*)d_ws;   // needs T*B*24*4 = 96 MiB of workspace

  // K1: 65536 tiles of 16 rows, 8 waves per 256-thread block
  k1_attn_proj<<<(NB / 16) * NT / 8, 256, 0, stream>>>(x, Wa, ba, Wl, bl, xz);
  // K2: 512 waves (8 batch rows x 4 gate groups each), 4 waves per block
  k2_lstm_tail<<<NB / 8 / 4, 128, 0, stream>>>(xz, Ul, W10, b10, Wo, bo, out);
}